// DigitCapsuleLayer_87522843559675
// MI455X (gfx1250) — compile-verified
//
#include <hip/hip_runtime.h>

typedef __attribute__((ext_vector_type(2))) float v2f;
typedef __attribute__((ext_vector_type(8))) float v8f;
typedef int v4i __attribute__((vector_size(16)));

#define B_      64
#define N_      10
#define IC_     6912
#define D_      16
#define NBT     4                 // batch tiles of 16 (waves per workgroup)
#define NWGJ    216               // workgroups along j
#define JCHUNK  (IC_ / NWGJ)      // 32 j's per workgroup
#define TILEF   (N_ * 256)        // floats per staged W tile (all 10 n, one j)
#define AOPS    5                 // async b128 ops per wave per tile (4*5*32*16B = 10KB)

#if defined(__has_builtin)
#if __has_builtin(__builtin_amdgcn_global_load_async_to_lds_b128)
#define ASYNC_LDS 1
typedef __attribute__((address_space(1))) v4i* gptr128;
typedef __attribute__((address_space(3))) v4i* lptr128;
#endif
#endif

#if defined(__has_builtin) && __has_builtin(__builtin_amdgcn_s_wait_asynccnt)
#define WAIT_ASYNC(n) __builtin_amdgcn_s_wait_asynccnt(n)
#else
#define WAIT_ASYNC(n) asm volatile("s_wait_asynccnt %0" ::"n"(n) : "memory")
#endif

// ---- zero the vsum accumulator -------------------------------------------
__global__ __launch_bounds__(128) void caps_zero(float* p, int n) {
  int i = blockIdx.x * blockDim.x + threadIdx.x;
  if (i < n) p[i] = 0.f;
}

// ---- one routing pass: recompute u_hat via WMMA, softmax over n, partial s
// part layout: [wgj*NBT + btile][n][o][bb]  (256 floats per (block,n))
__global__ __launch_bounds__(128) void caps_route_pass(
    const float* __restrict__ x, const float* __restrict__ W,
    const float* __restrict__ vsum, float* __restrict__ part)
{
  __shared__ float wlds[2][TILEF];   // double-buffered W tiles, natural [n][i][o]

  const int wgj   = blockIdx.x;
  const int btile = threadIdx.x >> 5;   // wave id == batch tile
  const int lane  = threadIdx.x & 31;
  const int h     = lane >> 4;          // half-wave
  const int bo    = lane & 15;          // bb for B-matrix lanes; o for A-matrix lanes
  const int b0    = btile * 16;
  const int jbase = wgj * JCHUNK;

  // ---- W tile staging: async Global->LDS (b128) with sync fallback ------
  auto stage = [&](int j, int bufi) {
#if defined(ASYNC_LDS)
#pragma unroll
    for (int t = 0; t < AOPS; ++t) {
      const int q = btile * (AOPS * 32) + t * 32 + lane;  // float4 index in tile
      const int n = q >> 6;
      const int r = (q & 63) << 2;                        // float offset in 256-block
      const float* g = W + (size_t)n * ((size_t)IC_ * 256) + (size_t)j * 256 + r;
      float* l = &wlds[bufi][q << 2];
      __builtin_amdgcn_global_load_async_to_lds_b128((gptr128)g, (lptr128)l, 0, 0);
    }
#else
    for (int f = threadIdx.x; f < TILEF; f += 128) {
      const int n = f >> 8, r = f & 255;
      wlds[bufi][f] = W[(size_t)n * ((size_t)IC_ * 256) + (size_t)j * 256 + r];
    }
#endif
  };

  // Preload routing vector sum: vv[n][k] = vsum[b0+bo][n][k+8h]
  float vv[N_][8];
  {
    const float* vsb = vsum + (size_t)(b0 + bo) * (N_ * D_);
#pragma unroll
    for (int n = 0; n < N_; ++n)
#pragma unroll
      for (int k = 0; k < 8; ++k)
        vv[n][k] = vsb[n * D_ + k + 8 * h];
  }

  const v8f vzero = {0.f, 0.f, 0.f, 0.f, 0.f, 0.f, 0.f, 0.f};
  v8f s_acc[N_];
#pragma unroll
  for (int n = 0; n < N_; ++n) s_acc[n] = vzero;

  // Prologue: stage first W tile, prefetch first x fragments
  stage(jbase, 0);
  v2f xb[4];
  {
    const float* xrow = x + ((size_t)(b0 + bo) * IC_ + jbase) * D_;
#pragma unroll
    for (int c = 0; c < 4; ++c) xb[c] = *(const v2f*)(xrow + 4 * c + 2 * h);
  }

  for (int jj = 0; jj < JCHUNK; ++jj) {
    const int cur = jj & 1;

    // prefetch next tile / next x while computing current
    v2f xb_next[4];
    if (jj + 1 < JCHUNK) {
      stage(jbase + jj + 1, cur ^ 1);
      const float* xrow = x + ((size_t)(b0 + bo) * IC_ + (jbase + jj + 1)) * D_;
#pragma unroll
      for (int c = 0; c < 4; ++c) xb_next[c] = *(const v2f*)(xrow + 4 * c + 2 * h);
#if defined(ASYNC_LDS)
      WAIT_ASYNC(AOPS);   // retire current buffer's ops; next buffer's stay in flight
#endif
    } else {
#if defined(ASYNC_LDS)
      WAIT_ASYNC(0);
#endif
    }
    __syncthreads();      // all waves' staging of buf[cur] visible

    const float* wbuf = &wlds[cur][0];
    float blog[N_];
    v8f uh[N_];
#pragma unroll
    for (int n = 0; n < N_; ++n) {
      // A-fragments: A[M=o][K=i] = W[n][i][o]; lane row o=bo, strided b32 pair
      const float* wcol = wbuf + n * 256 + bo;
      v8f u = vzero;
#pragma unroll
      for (int c = 0; c < 4; ++c) {
        const int i0 = 4 * c + 2 * h;
        v2f wa = {wcol[i0 * 16], wcol[i0 * 16 + 16]};
        u = __builtin_amdgcn_wmma_f32_16x16x4_f32(false, wa, false, xb[c],
                                                  (short)0, u, false, false);
      }
      uh[n] = u;

      // routing logit: blog[bb][n] = sum_o u_hat[o][bb] * vsum[n][o]
      float bl = 0.f;
#pragma unroll
      for (int k = 0; k < 8; ++k) bl += u[k] * vv[n][k];
      bl += __shfl_xor(bl, 16, 32);   // combine o-halves
      blog[n] = bl;
    }

    // softmax over the 10 output capsules (per bb)
    float m = blog[0];
#pragma unroll
    for (int n = 1; n < N_; ++n) m = fmaxf(m, blog[n]);
    float e[N_];
    float den = 0.f;
#pragma unroll
    for (int n = 0; n < N_; ++n) { e[n] = __expf(blog[n] - m); den += e[n]; }
    const float inv = 1.f / den;

#pragma unroll
    for (int n = 0; n < N_; ++n) {
      const float cc = e[n] * inv;
#pragma unroll
      for (int k = 0; k < 8; ++k) s_acc[n][k] += cc * uh[n][k];
    }

#pragma unroll
    for (int c = 0; c < 4; ++c) xb[c] = xb_next[c];

    __syncthreads();      // buf[cur] reads done before it is overwritten (jj+2)
  }

  // coalesced partial-s store: 16 lanes (bb) contiguous per (n, o)
  float* pout = part + (size_t)(wgj * NBT + btile) * (N_ * 256);
#pragma unroll
  for (int n = 0; n < N_; ++n)
#pragma unroll
    for (int k = 0; k < 8; ++k)
      pout[n * 256 + (k + 8 * h) * 16 + bo] = s_acc[n][k];
}

// ---- reduce partial s over workgroups, squash, update vsum, emit v -------
__global__ __launch_bounds__(128) void caps_reduce_squash(
    const float* __restrict__ part, float* __restrict__ vsum,
    float* __restrict__ out, int last)
{
  const int t = threadIdx.x;
  const int o = t & 15;
  const int p = blockIdx.x * 8 + (t >> 4);   // (b,n) pair, 640 total
  const int b = p / 10, n = p - 10 * (p / 10);
  const int btile = b >> 4, bb = b & 15;

  float s = 0.f;
  const float* pp = part + (size_t)btile * (N_ * 256) + n * 256 + o * 16 + bb;
  for (int wg = 0; wg < NWGJ; ++wg)
    s += pp[(size_t)wg * NBT * (N_ * 256)];

  float s2 = s * s;
#pragma unroll
  for (int off = 1; off < 16; off <<= 1) s2 += __shfl_xor(s2, off, 16);

  const float scale = (s2 / (1.f + s2)) * rsqrtf(s2 + 1e-7f);
  const float v = scale * s;
  const size_t oidx = ((size_t)b * N_ + n) * D_ + o;
  vsum[oidx] += v;
  if (last) out[oidx] = v;
}

extern "C" void kernel_launch(void* const* d_in, const int* in_sizes, int n_in,
                              void* d_out, int out_size, void* d_ws, size_t ws_size,
                              hipStream_t stream) {
  const float* x = (const float*)d_in[0];   // [64, 6912, 16]
  const float* W = (const float*)d_in[1];   // [10, 6912, 16, 16]
  float* out  = (float*)d_out;              // [64, 10, 16]
  float* vsum = (float*)d_ws;               // 10240 floats
  float* part = vsum + B_ * N_ * D_;        // NWGJ*NBT*N_*256 floats (~8.85 MB)

  caps_zero<<<(B_ * N_ * D_ + 127) / 128, 128, 0, stream>>>(vsum, B_ * N_ * D_);

  for (int it = 0; it < 3; ++it) {
    caps_route_pass<<<NWGJ, 128, 0, stream>>>(x, W, vsum, part);
    caps_reduce_squash<<<(B_ * N_) / 8, 128, 0, stream>>>(part, vsum, out, it == 2 ? 1 : 0);
  }
}